// DecoderLayer_16097537426031
// MI455X (gfx1250) — compile-verified
//
#include <hip/hip_runtime.h>

// Problem constants
#define NN    3000            // nodes
#define NP    3072            // nodes padded to 128-multiple for GEMM tiling
#define BATCH 16
#define CCH   64              // channels
#define TT    12              // time
#define MC    12288           // GEMM columns = BATCH*TT*CCH
#define NTB   36000           // N*T        (per batch, per channel)
#define BCH   2304000         // C*N*T      (per batch)
#define TOT   36864000        // B*C*N*T
#define ALPHA 0.05f
#define LNEPS 1e-5f

typedef float v2f __attribute__((ext_vector_type(2)));
typedef float v8f __attribute__((ext_vector_type(8)));

// ---------------------------------------------------------------- adjacency
__global__ __launch_bounds__(256) void k_adj_fill(const float* __restrict__ adp,
                                                  float* __restrict__ M) {
  size_t i = (size_t)blockIdx.x * 256 + threadIdx.x;   // < NP*NP
  int v = (int)(i / NP), w = (int)(i % NP);
  float val = 0.f;
  if (v < NN && w < NN) {
    val = adp[(size_t)v * NN + w];
    if (v == w) val += 1.f;
  }
  M[i] = val;
}

__global__ __launch_bounds__(256) void k_row_inv(const float* __restrict__ M,
                                                 float* __restrict__ invr) {
  __shared__ float red[256];
  int v = blockIdx.x;
  float s = 0.f;
  for (int w = threadIdx.x; w < NP; w += 256) s += M[(size_t)v * NP + w];
  red[threadIdx.x] = s;
  __syncthreads();
  for (int st = 128; st > 0; st >>= 1) {
    if ((int)threadIdx.x < st) red[threadIdx.x] += red[threadIdx.x + st];
    __syncthreads();
  }
  if (threadIdx.x == 0) invr[v] = red[0] > 0.f ? 1.f / red[0] : 0.f;
}

__global__ __launch_bounds__(256) void k_col_inv(const float* __restrict__ M,
                                                 float* __restrict__ invc) {
  int v = blockIdx.x * 256 + threadIdx.x;
  float s = 0.f;
  for (int w = 0; w < NP; ++w) s += M[(size_t)w * NP + v];
  invc[v] = s > 0.f ? 1.f / s : 0.f;
}

// ------------------------------------------------- x[b,c,n,t] -> Xp[n,(b,t,c)]
__global__ __launch_bounds__(256) void k_transpose(const float* __restrict__ x,
                                                   float* __restrict__ Xp) {
  __shared__ float lds[16 * TT * 65];
  int b = blockIdx.y, n0 = blockIdx.x * 16;
  for (int i = threadIdx.x; i < 16 * CCH * TT; i += 256) {
    int c = i / (16 * TT);
    int rem = i % (16 * TT);
    int nl = rem / TT, t = rem % TT;
    int n = n0 + nl;
    float v = 0.f;
    if (n < NN) v = x[(((size_t)b * CCH + c) * NN + n) * TT + t];
    lds[(nl * TT + t) * 65 + c] = v;
  }
  __syncthreads();
  for (int i = threadIdx.x; i < 16 * CCH * TT; i += 256) {
    int c = i & 63;
    int t = (i >> 6) % TT;
    int nl = i / (CCH * TT);
    Xp[(size_t)(n0 + nl) * MC + (size_t)b * (TT * CCH) + t * CCH + c] =
        lds[(nl * TT + t) * 65 + c];
  }
}

// -------------------------------------------- node-propagation GEMM (WMMA f32)
// Out[v,m] = ALPHA*Xa[v,m] + (1-ALPHA)*invs[v] * sum_w Aop[v,w]*Bm[w,m]
// Aop = A (transA=0) or A^T (transA=1).  WG: 8 wave32 waves, tile 128v x 64m.
__global__ __launch_bounds__(256) void k_prop(const float* __restrict__ A,
                                              const float* __restrict__ Bm,
                                              const float* __restrict__ Xa,
                                              float* __restrict__ Out,
                                              const float* __restrict__ invs,
                                              int transA) {
  __shared__ float lA[128 * 20];
  __shared__ float lB[16 * 68];
  const int tid = threadIdx.x;
  const size_t mblk = (size_t)blockIdx.x * 64;
  const int vblk = blockIdx.y * 128;
  const int wid = tid >> 5, lane = tid & 31;
  const int wv = wid & 3, wm = wid >> 2;     // 4 waves along v, 2 along m
  const int lrow = lane & 15, lhi = lane >> 4;

  v8f zero = {0.f, 0.f, 0.f, 0.f, 0.f, 0.f, 0.f, 0.f};
  v8f acc[2][2];
  acc[0][0] = zero; acc[0][1] = zero; acc[1][0] = zero; acc[1][1] = zero;

  for (int k0 = 0; k0 < NP; k0 += 16) {
    // ---- stage A tile (128 x 16) ----
    if (!transA) {
      int row = tid >> 1, col = (tid & 1) * 8;
      const float4* s = (const float4*)(A + (size_t)(vblk + row) * NP + k0 + col);
      float4 p0 = s[0], p1 = s[1];
      float* d = &lA[row * 20 + col];
      d[0] = p0.x; d[1] = p0.y; d[2] = p0.z; d[3] = p0.w;
      d[4] = p1.x; d[5] = p1.y; d[6] = p1.z; d[7] = p1.w;
    } else {
      int v = tid & 127, kh = tid >> 7;
#pragma unroll
      for (int kk = 0; kk < 8; ++kk) {
        int k = kk * 2 + kh;
        lA[v * 20 + k] = A[(size_t)(k0 + k) * NP + vblk + v];
      }
    }
    // ---- stage B tile (16 x 64) ----
    {
      int k = tid >> 4, m4 = (tid & 15) * 4;
      float4 p = *(const float4*)(Bm + (size_t)(k0 + k) * MC + mblk + m4);
      float* d = &lB[k * 68 + m4];
      d[0] = p.x; d[1] = p.y; d[2] = p.z; d[3] = p.w;
      if (k0 + 16 < NP)  // pull next B tile toward the caches
        __builtin_prefetch(Bm + (size_t)(k0 + 16 + k) * MC + mblk + m4, 0, 3);
    }
    __syncthreads();

#pragma unroll
    for (int kc = 0; kc < 4; ++kc) {
      // A frag: lane holds row M=lrow, K = kc*4 + 2*lhi + j
      int acol = kc * 4 + 2 * lhi;
      int ar0 = (wv * 32 + lrow) * 20 + acol;
      int ar1 = (wv * 32 + 16 + lrow) * 20 + acol;
      v2f fa0, fa1, fb0, fb1;
      fa0.x = lA[ar0]; fa0.y = lA[ar0 + 1];
      fa1.x = lA[ar1]; fa1.y = lA[ar1 + 1];
      // B frag: lane holds col N=lrow, K = kc*4 + 2*lhi + j
      int brow = (kc * 4 + 2 * lhi) * 68;
      fb0.x = lB[brow + wm * 32 + lrow];
      fb0.y = lB[brow + 68 + wm * 32 + lrow];
      fb1.x = lB[brow + wm * 32 + 16 + lrow];
      fb1.y = lB[brow + 68 + wm * 32 + 16 + lrow];
      acc[0][0] = __builtin_amdgcn_wmma_f32_16x16x4_f32(false, fa0, false, fb0, (short)0, acc[0][0], false, false);
      acc[0][1] = __builtin_amdgcn_wmma_f32_16x16x4_f32(false, fa0, false, fb1, (short)0, acc[0][1], false, false);
      acc[1][0] = __builtin_amdgcn_wmma_f32_16x16x4_f32(false, fa1, false, fb0, (short)0, acc[1][0], false, false);
      acc[1][1] = __builtin_amdgcn_wmma_f32_16x16x4_f32(false, fa1, false, fb1, (short)0, acc[1][1], false, false);
    }
    __syncthreads();
  }

  // fused mixprop epilogue: h = alpha*x + (1-alpha)*normalized propagation
#pragma unroll
  for (int sv = 0; sv < 2; ++sv)
#pragma unroll
    for (int sm = 0; sm < 2; ++sm)
#pragma unroll
      for (int r = 0; r < 8; ++r) {
        int v = vblk + wv * 32 + sv * 16 + r + 8 * lhi;
        size_t idx = (size_t)v * MC + mblk + wm * 32 + sm * 16 + lrow;
        Out[idx] = ALPHA * Xa[idx] + (1.f - ALPHA) * acc[sv][sm][r] * invs[v];
      }
}

// ----------------------------- channel-mix 1x1 conv over 192 hop-channels (WMMA)
// out[b,o,n,t] (+)= sum_{hop,c} W[o, hop*64+c] * Hhop[n,(b,t,c)]  (+bias+residual)
__global__ __launch_bounds__(256) void k_mix(const float* __restrict__ W,
                                             const float* __restrict__ H0,
                                             const float* __restrict__ H1,
                                             const float* __restrict__ H2,
                                             const float* __restrict__ bw1,
                                             const float* __restrict__ bw2,
                                             const float* __restrict__ xres,
                                             float* __restrict__ out,
                                             int addBiasRes) {
  __shared__ float lB[16 * 68];
  const int tid = threadIdx.x;
  const size_t mblk = (size_t)blockIdx.x * 64;   // 64 (b,n,t) columns per WG
  const int wid = tid >> 5, lane = tid & 31;
  const int wo = wid & 3, wm2 = wid >> 2;        // 4 waves on o, 2 on m
  const int lrow = lane & 15, lhi = lane >> 4;

  // per-thread B-stage column address (same for all K iterations)
  int colL = tid >> 2;
  int j4 = (tid & 3) * 4;
  size_t col_off;
  {
    int mm = (int)mblk + colL;
    int b = mm / NTB;
    int loc = mm - b * NTB;
    int n = loc / TT, t = loc - n * TT;
    col_off = (size_t)n * MC + (size_t)b * (TT * CCH) + t * CCH;
  }

  v8f zero = {0.f, 0.f, 0.f, 0.f, 0.f, 0.f, 0.f, 0.f};
  v8f acc[2];
  acc[0] = zero; acc[1] = zero;

  for (int ki = 0; ki < 12; ++ki) {
    int kbase = ki * 16;
    const float* Hs = (kbase < 64) ? H0 : (kbase < 128 ? H1 : H2);
    int c0 = kbase & 63;
    float4 p = *(const float4*)(Hs + col_off + c0 + j4);
    lB[(j4 + 0) * 68 + colL] = p.x;
    lB[(j4 + 1) * 68 + colL] = p.y;
    lB[(j4 + 2) * 68 + colL] = p.z;
    lB[(j4 + 3) * 68 + colL] = p.w;
    __syncthreads();
#pragma unroll
    for (int kc = 0; kc < 4; ++kc) {
      int acol = kbase + kc * 4 + 2 * lhi;
      int o = wo * 16 + lrow;
      v2f fa, fb0, fb1;
      fa.x = W[o * 192 + acol];
      fa.y = W[o * 192 + acol + 1];
      int brow = (kc * 4 + 2 * lhi) * 68;
      fb0.x = lB[brow + wm2 * 32 + lrow];
      fb0.y = lB[brow + 68 + wm2 * 32 + lrow];
      fb1.x = lB[brow + wm2 * 32 + 16 + lrow];
      fb1.y = lB[brow + 68 + wm2 * 32 + 16 + lrow];
      acc[0] = __builtin_amdgcn_wmma_f32_16x16x4_f32(false, fa, false, fb0, (short)0, acc[0], false, false);
      acc[1] = __builtin_amdgcn_wmma_f32_16x16x4_f32(false, fa, false, fb1, (short)0, acc[1], false, false);
    }
    __syncthreads();
  }

#pragma unroll
  for (int sm = 0; sm < 2; ++sm) {
    int mm = (int)mblk + wm2 * 32 + sm * 16 + lrow;
    int b = mm / NTB;
    int loc = mm - b * NTB;
#pragma unroll
    for (int r = 0; r < 8; ++r) {
      int o = wo * 16 + r + 8 * lhi;
      size_t addr = (size_t)b * BCH + (size_t)o * NTB + loc;
      float v = acc[sm][r];
      if (addBiasRes)
        out[addr] = v + bw1[o] + bw2[o] + xres[addr];
      else
        out[addr] += v;
    }
  }
}

// --------------------------------------------------------------- LayerNorm
__global__ __launch_bounds__(256) void k_ln_part(const float* __restrict__ out,
                                                 float* __restrict__ part) {
  __shared__ float rs[256], rq[256];
  int b = blockIdx.y, chunk = blockIdx.x;
  size_t base = (size_t)b * BCH + (size_t)chunk * NTB;
  float s = 0.f, q = 0.f;
  for (int i = threadIdx.x; i < NTB; i += 256) {
    float v = out[base + i];
    s += v;
    q += v * v;
  }
  rs[threadIdx.x] = s; rq[threadIdx.x] = q;
  __syncthreads();
  for (int st = 128; st > 0; st >>= 1) {
    if ((int)threadIdx.x < st) {
      rs[threadIdx.x] += rs[threadIdx.x + st];
      rq[threadIdx.x] += rq[threadIdx.x + st];
    }
    __syncthreads();
  }
  if (threadIdx.x == 0) {
    part[(b * 64 + chunk) * 2 + 0] = rs[0];
    part[(b * 64 + chunk) * 2 + 1] = rq[0];
  }
}

__global__ __launch_bounds__(64) void k_ln_final(const float* __restrict__ part,
                                                 float* __restrict__ stats) {
  __shared__ float rs[64], rq[64];
  int b = blockIdx.x;
  rs[threadIdx.x] = part[(b * 64 + threadIdx.x) * 2 + 0];
  rq[threadIdx.x] = part[(b * 64 + threadIdx.x) * 2 + 1];
  __syncthreads();
  for (int st = 32; st > 0; st >>= 1) {
    if ((int)threadIdx.x < st) {
      rs[threadIdx.x] += rs[threadIdx.x + st];
      rq[threadIdx.x] += rq[threadIdx.x + st];
    }
    __syncthreads();
  }
  if (threadIdx.x == 0) {
    float mu = rs[0] / (float)BCH;
    float var = rq[0] / (float)BCH - mu * mu;
    stats[b * 2 + 0] = mu;
    stats[b * 2 + 1] = rsqrtf(var + LNEPS);
  }
}

__global__ __launch_bounds__(256) void k_ln_apply(float* __restrict__ out,
                                                  const float* __restrict__ stats,
                                                  const float* __restrict__ lnw,
                                                  const float* __restrict__ lnb,
                                                  const int* __restrict__ idx) {
  int i = blockIdx.x * 256 + threadIdx.x;     // < TOT (fits int32)
  int b = i / BCH;
  int rem = i - b * BCH;
  int c = rem / NTB;
  int r2 = rem - c * NTB;
  int n = r2 / TT, t = r2 - n * TT;
  int li = c * NTB + idx[n] * TT + t;
  float mu = stats[b * 2 + 0], rsg = stats[b * 2 + 1];
  out[i] = (out[i] - mu) * rsg * lnw[li] + lnb[li];
}

// ------------------------------------------------------------------- launch
extern "C" void kernel_launch(void* const* d_in, const int* in_sizes, int n_in,
                              void* d_out, int out_size, void* d_ws, size_t ws_size,
                              hipStream_t stream) {
  (void)in_sizes; (void)n_in; (void)out_size; (void)ws_size;
  const float* x   = (const float*)d_in[0];
  const float* adp = (const float*)d_in[1];
  const float* W1  = (const float*)d_in[2];
  const float* b1  = (const float*)d_in[3];
  const float* W2  = (const float*)d_in[4];
  const float* b2  = (const float*)d_in[5];
  const float* lnw = (const float*)d_in[6];
  const float* lnb = (const float*)d_in[7];
  const int*   idx = (const int*)d_in[8];
  float* out = (float*)d_out;

  char* ws = (char*)d_ws;
  size_t off = 0;
  auto alloc = [&](size_t bytes) {
    char* p = ws + off;
    off += (bytes + 255) & ~(size_t)255;
    return p;
  };
  float* adjM = (float*)alloc((size_t)NP * NP * sizeof(float));   //  37.7 MB
  float* invr = (float*)alloc((size_t)NP * sizeof(float));
  float* invc = (float*)alloc((size_t)NP * sizeof(float));
  float* Xp   = (float*)alloc((size_t)NP * MC * sizeof(float));   // 151 MB
  float* H1   = (float*)alloc((size_t)NP * MC * sizeof(float));   // 151 MB
  float* H2   = (float*)alloc((size_t)NP * MC * sizeof(float));   // 151 MB
  float* part = (float*)alloc((size_t)BATCH * 64 * 2 * sizeof(float));
  float* stats= (float*)alloc((size_t)BATCH * 2 * sizeof(float));

  // adjacency (+self loops), row/col normalizers
  k_adj_fill<<<(NP * NP) / 256, 256, 0, stream>>>(adp, adjM);
  k_row_inv<<<NP, 256, 0, stream>>>(adjM, invr);
  k_col_inv<<<NP / 256, 256, 0, stream>>>(adjM, invc);

  // x -> node-major layout
  k_transpose<<<dim3(NP / 16, BATCH), 256, 0, stream>>>(x, Xp);

  dim3 pg(MC / 64, NP / 128);
  // forward direction (row-normalized A)
  k_prop<<<pg, 256, 0, stream>>>(adjM, Xp, Xp, H1, invr, 0);
  k_prop<<<pg, 256, 0, stream>>>(adjM, H1, Xp, H2, invr, 0);
  k_mix<<<(BATCH * NTB) / 64, 256, 0, stream>>>(W1, Xp, H1, H2, b1, b2, x, out, 1);
  // backward direction (A^T, column-normalized)
  k_prop<<<pg, 256, 0, stream>>>(adjM, Xp, Xp, H1, invc, 1);
  k_prop<<<pg, 256, 0, stream>>>(adjM, H1, Xp, H2, invc, 1);
  k_mix<<<(BATCH * NTB) / 64, 256, 0, stream>>>(W2, Xp, H1, H2, b1, b2, x, out, 0);

  // LayerNorm over (C,N,T) per batch
  k_ln_part<<<dim3(64, BATCH), 256, 0, stream>>>(out, part);
  k_ln_final<<<BATCH, 64, 0, stream>>>(part, stats);
  k_ln_apply<<<TOT / 256, 256, 0, stream>>>(out, stats, lnw, lnb, idx);
}